// Aggregator_12524124636045
// MI455X (gfx1250) — compile-verified
//
#include <hip/hip_runtime.h>

#define D 128
#define LEAKY 0.01f
#define ASTR 132   // padded LDS row stride (floats): bank-conflict-free for WMMA A reads

typedef float v2f __attribute__((ext_vector_type(2)));
typedef float v8f __attribute__((ext_vector_type(8)));

// ---------------- Kernel 1: zero the H accumulator (workspace) ----------------
__global__ void zero_H_kernel(float4* __restrict__ H4, int n4) {
    int i = blockIdx.x * blockDim.x + threadIdx.x;
    if (i < n4) {
        float4 z; z.x = 0.f; z.y = 0.f; z.z = 0.f; z.w = 0.f;
        H4[i] = z;
    }
}

// ---------------- Kernel 2: edge scatter-add, one wave32 per edge ----------------
// H[dst[e]][:] += E[src[e]][:]   (128 floats per edge; lane handles 4 channels)
// Edge indices are wave-uniform: force them scalar so src/dst fetches become
// s_load (KMcnt) and the vector pipe only carries the gather + atomics.
__global__ __launch_bounds__(256) void edge_scatter_kernel(
    const float* __restrict__ E,
    const int* __restrict__ src,
    const int* __restrict__ dst,
    float* __restrict__ H,
    int n_edges)
{
    int e = blockIdx.x * 8 + (threadIdx.x >> 5);   // wave id = edge id
    int lane = threadIdx.x & 31;
    if (e >= n_edges) return;
    e = __builtin_amdgcn_readfirstlane(e);
    const int s = __builtin_amdgcn_readfirstlane(src[e]);
    const int d = __builtin_amdgcn_readfirstlane(dst[e]);
    const float4 v = *(const float4*)(E + (size_t)s * D + lane * 4);
    float* o = H + (size_t)d * D + lane * 4;
    // hardware global_atomic_add_f32 (no-return), avoids CAS loops
    unsafeAtomicAdd(o + 0, v.x);
    unsafeAtomicAdd(o + 1, v.y);
    unsafeAtomicAdd(o + 2, v.z);
    unsafeAtomicAdd(o + 3, v.w);
}

// ---------------- Kernel 3: fused dual GEMM + bias + LeakyReLU + add ----------------
// out = leaky(H @ W1^T + b1) + leaky((E*H) @ W2^T + b2)
// Block = 16 nodes x 128 outputs; 8 waves, each wave owns a 16-wide N tile,
// runs both branches with fp32 WMMA (V_WMMA_F32_16X16X4_F32), K looped 128/4.
__global__ __launch_bounds__(256) void gemm_epilogue_kernel(
    const float* __restrict__ H,
    const float* __restrict__ E,
    const float* __restrict__ W1,
    const float* __restrict__ b1,
    const float* __restrict__ W2,
    const float* __restrict__ b2,
    float* __restrict__ out,
    int n_nodes)
{
    __shared__ float sH[16 * ASTR];    // H tile,   16 x 128 (padded)
    __shared__ float sEH[16 * ASTR];   // E*H tile, 16 x 128 (padded)

    const int t = threadIdx.x;
    const int node0 = blockIdx.x * 16;

    // Cooperative stage: 256 threads x 8 floats = 16x128 tile, two float4 each.
    {
        int row = t >> 4;            // 0..15
        int col = (t & 15) * 8;      // 0..120
        int r = node0 + row;
        if (r >= n_nodes) r = n_nodes - 1;     // clamp (tail-safe; 50000%16==0 anyway)
        const float4 h0 = *(const float4*)(H + (size_t)r * D + col);
        const float4 h1 = *(const float4*)(H + (size_t)r * D + col + 4);
        const float4 e0 = *(const float4*)(E + (size_t)r * D + col);
        const float4 e1 = *(const float4*)(E + (size_t)r * D + col + 4);
        *(float4*)(sH + row * ASTR + col)     = h0;
        *(float4*)(sH + row * ASTR + col + 4) = h1;
        float4 p0, p1;
        p0.x = h0.x * e0.x; p0.y = h0.y * e0.y; p0.z = h0.z * e0.z; p0.w = h0.w * e0.w;
        p1.x = h1.x * e1.x; p1.y = h1.y * e1.y; p1.z = h1.z * e1.z; p1.w = h1.w * e1.w;
        *(float4*)(sEH + row * ASTR + col)     = p0;
        *(float4*)(sEH + row * ASTR + col + 4) = p1;
    }
    __syncthreads();

    const int wave  = t >> 5;          // 0..7 -> N tile index
    const int lane  = t & 31;
    const int mrow  = lane & 15;       // A: M row this lane holds / B,C,D: N column
    const int khalf = (lane >> 4) * 2; // upper half-wave holds K+2,K+3 (ISA A/B layout)
    const int ncol  = wave * 16 + mrow;            // global output column
    const float* w1p = W1 + (size_t)ncol * D;      // B[k][n] = W1[n][k] -> contiguous in k
    const float* w2p = W2 + (size_t)ncol * D;

    v8f c1 = {};
    v8f c2 = {};
    #pragma unroll 4
    for (int k = 0; k < D; k += 4) {
        const int kk = k + khalf;
        v2f a1  = *(const v2f*)(sH  + mrow * ASTR + kk);
        v2f a2  = *(const v2f*)(sEH + mrow * ASTR + kk);
        v2f bb1 = *(const v2f*)(w1p + kk);
        v2f bb2 = *(const v2f*)(w2p + kk);
        // D = A x B + C, fp32 in / fp32 accumulate (matches fp32 reference)
        c1 = __builtin_amdgcn_wmma_f32_16x16x4_f32(
                 false, a1, false, bb1, (short)0, c1, false, false);
        c2 = __builtin_amdgcn_wmma_f32_16x16x4_f32(
                 false, a2, false, bb2, (short)0, c2, false, false);
    }

    // Epilogue. C/D layout: VGPR r -> M = r (+8 for upper half-wave), N = lane&15.
    const float bias1 = b1[ncol];
    const float bias2 = b2[ncol];
    const int mbase = (lane >> 4) * 8;

    float res[8];
    #pragma unroll
    for (int r = 0; r < 8; ++r) {
        float x1 = c1[r] + bias1; x1 = (x1 > 0.f) ? x1 : LEAKY * x1;
        float x2 = c2[r] + bias2; x2 = (x2 > 0.f) ? x2 : LEAKY * x2;
        res[r] = x1 + x2;
    }

    float* op = out + (size_t)(node0 + mbase) * D + ncol;
    if (node0 + 16 <= n_nodes) {
        // fast path: full tile, unguarded strided stores (clause-friendly)
        #pragma unroll
        for (int r = 0; r < 8; ++r) op[(size_t)r * D] = res[r];
    } else {
        #pragma unroll
        for (int r = 0; r < 8; ++r)
            if (node0 + mbase + r < n_nodes) op[(size_t)r * D] = res[r];
    }
}

extern "C" void kernel_launch(void* const* d_in, const int* in_sizes, int n_in,
                              void* d_out, int out_size, void* d_ws, size_t ws_size,
                              hipStream_t stream) {
    const float* E  = (const float*)d_in[0];   // entity_embed [N, 128]
    const float* W1 = (const float*)d_in[1];   // [128, 128]
    const float* b1 = (const float*)d_in[2];   // [128]
    const float* W2 = (const float*)d_in[3];   // [128, 128]
    const float* b2 = (const float*)d_in[4];   // [128]
    const int* src  = (const int*)d_in[5];     // [n_edges]
    const int* dst  = (const int*)d_in[6];     // [n_edges]
    float* out = (float*)d_out;
    float* H   = (float*)d_ws;                 // [N, 128] fp32 accumulator (25.6 MB)

    const int n_nodes = in_sizes[0] / D;
    const int n_edges = in_sizes[5];

    // 1) H = 0
    const int n4 = (n_nodes * D) / 4;
    zero_H_kernel<<<(n4 + 255) / 256, 256, 0, stream>>>((float4*)H, n4);

    // 2) scatter-add edges (one wave per edge, 8 waves per block)
    const int sblk = (n_edges + 7) / 8;
    edge_scatter_kernel<<<sblk, 256, 0, stream>>>(E, src, dst, H, n_edges);

    // 3) fused dual WMMA GEMM + epilogue
    const int gblk = (n_nodes + 15) / 16;
    gemm_epilogue_kernel<<<gblk, 256, 0, stream>>>(H, E, W1, b1, W2, b2, out, n_nodes);
}